// AttentionModelBase_55954833932722
// MI455X (gfx1250) — compile-verified
//
#include <hip/hip_runtime.h>
#include <math.h>

// ---------------------------------------------------------------------------
// One decode step of AttentionModelBase, fused into a single kernel.
// Grid = B blocks (one per batch row), 128 threads (= 4 wave32) per block.
//
// Algebraic refactor (exact up to FP reassociation):
//   scores[h,n] = emb[n] . u_h,   u_h = (1/sqrt(dh)) * Wk[:,h*16:+16] @ q_h
//   a_h         = sum_n p[h,n] * emb[n]       (online/flash softmax)
//   heads[h*16+j] = (a_h / l_h) . Wv[:, h*16+j]
//   logits[n]   = emb[n] . wvec,  wvec = (1/sqrt(D)) * Wl @ glimpse
// => no [B,N,3D] intermediate, ~2.5 GFLOP total, 3 streaming sweeps of the
//    per-b 512KB embedding slice (sweeps 2-3 mostly L2-resident on MI455X).
//
// CDNA5 specifics:
//  - V_WMMA_F32_16X16X4_F32 for the score tile and the P@emb accumulation
//  - TENSOR_LOAD_TO_LDS (TDM) double-buffered staging of emb tiles, with the
//    descriptor pad feature (4 dwords per 128) producing the 132-float row
//    stride that keeps the column-major A-fragment LDS reads conflict-free;
//    tensor_dim1 = rows-remaining gives hardware zero-fill of the last tile.
//  - s_wait_tensorcnt for completion, global_prefetch_b8 in the logits sweep.
// ---------------------------------------------------------------------------

typedef __attribute__((ext_vector_type(2))) float        v2f;
typedef __attribute__((ext_vector_type(8))) float        v8f;
typedef __attribute__((ext_vector_type(4))) unsigned int v4u;
typedef __attribute__((ext_vector_type(8))) int          v8i;
typedef __attribute__((ext_vector_type(4))) int          v4i;

#define B_    512
#define N_    1000
#define D_    128
#define H_    8
#define NTILE 63            // ceil(N_/16); last tile has 8 valid rows
#define TROW  132           // padded LDS row stride (128 + 4 dwords TDM pad)
#define INVSQRT_D 0.08838834764831845f

__device__ __forceinline__ v8f wmma_f32(v2f a, v2f b, v8f c) {
  // 8 args: (neg_a, A, neg_b, B, c_mod, C, reuse_a, reuse_b)
  return __builtin_amdgcn_wmma_f32_16x16x4_f32(false, a, false, b, (short)0, c,
                                               false, false);
}

// Issue one TDM load of a 16x128 f32 tile (row-padded to 132 dwords in LDS).
// rows_left clamps tensor_dim1 so OOB rows of the last tile zero-fill.
__device__ __forceinline__ void tdm_load_tile(const float* gsrc,
                                              unsigned lds_off, int rows_left) {
  const unsigned long long ga = (unsigned long long)(uintptr_t)gsrc;
  v4u g0;
  g0.x = 1u;                                   // count=1, user descriptor
  g0.y = lds_off;                              // lds_addr (bytes)
  g0.z = (unsigned)ga;                         // global_addr[31:0]
  g0.w = (unsigned)((ga >> 32) & 0x01FFFFFFull) | (2u << 30);  // addr hi|type=2
  v8i g1;
  g1[0] = (int)((2u << 16)       // data_size = 4 bytes
              | (1u << 20)       // pad_enable
              | (6u << 22)       // pad_interval: every 128 dwords
              | (3u << 25));     // pad_amount: 4 dwords  -> 132-dword rows
  g1[1] = (int)(128u << 16);     // tensor_dim0 = 128 (bits 63:48)
  g1[2] = (int)((unsigned)rows_left << 16);  // tensor_dim1[15:0] (bits 111:80 low half)
  g1[3] = (int)(128u << 16);     // tile_dim0 = 128 (bits 127:112)
  g1[4] = 16;                    // tile_dim1 = 16 (bits 143:128)
  g1[5] = 128;                   // tensor_dim0_stride = 128 (bits 191:160)
  g1[6] = 0;
  g1[7] = 0;
  const v4i z4 = {0, 0, 0, 0};             // groups 2/3 unused (2D tensor)
  const v8i z8 = {0, 0, 0, 0, 0, 0, 0, 0}; // trailing group (clang-23 form)
  __builtin_amdgcn_tensor_load_to_lds(g0, g1, z4, z4, z8, 0);
}

__global__ __launch_bounds__(128)
void attn_decode_step_kernel(const float* __restrict__ emb,     // [B,N,D]
                             const int*   __restrict__ mask,    // [B,1,N]
                             const float* __restrict__ W_node,  // [D,3D]
                             const float* __restrict__ W_fixed, // [D,D]
                             const float* __restrict__ W_step,  // [2D,D]
                             const float* __restrict__ W_out,   // [D,D]
                             const float* __restrict__ W_ph,    // [2D]
                             float* __restrict__ out)           // [B,1,N]
{
  __shared__ float ge[D_];              // graph embedding (mean)
  __shared__ float query[D_];
  __shared__ float Ulds[D_][16];        // u_h vectors, cols 8..15 = 0
  __shared__ float Temb[2][16][TROW];   // double-buffered emb tiles (TDM dest)
  __shared__ float Slds[16][17];        // score tile [n_local][h]
  __shared__ float Plds[16][17];        // softmax-weight tile, cols>=8 stay 0
  __shared__ float scaleLds[16];        // per-head rescale (rows 8..15 = 1.0)
  __shared__ float lLds[H_];
  __shared__ float AccLds[16][D_];      // unnormalized heads-accumulator dump
  __shared__ float headsLds[D_];
  __shared__ float glimpseLds[D_];
  __shared__ float wvec[D_];
  __shared__ float logitsLds[N_];
  __shared__ float red[128];

  const int b    = blockIdx.x;
  const int t    = threadIdx.x;
  const int lane = t & 31;
  const int wave = t >> 5;

  const float* embB  = emb  + (size_t)b * N_ * D_;
  const int*   maskB = mask + (size_t)b * N_;

  // ---------------- Phase 0: graph mean -> query -> u_h vectors -------------
  {
    float s0 = 0.f, s1 = 0.f, s2 = 0.f, s3 = 0.f;
    for (int n = 0; n < N_; n += 4) {
      s0 += embB[(n + 0) * D_ + t];
      s1 += embB[(n + 1) * D_ + t];
      s2 += embB[(n + 2) * D_ + t];
      s3 += embB[(n + 3) * D_ + t];
    }
    ge[t] = (s0 + s1 + s2 + s3) * (1.0f / N_);
  }
  __syncthreads();
  {
    float q = 0.f;
    for (int d = 0; d < D_; ++d)      q += ge[d] * W_fixed[d * D_ + t];
    for (int k = 0; k < 2 * D_; ++k)  q += W_ph[k] * W_step[k * D_ + t];
    query[t] = q;
  }
  __syncthreads();
  {
    // thread t owns row d=t of U;  u[d][h] = 0.25 * sum_j Wk[d,h*16+j]*q[h*16+j]
    const float* WkRow = W_node + (size_t)t * (3 * D_);
    for (int h = 0; h < 16; ++h) {
      float u = 0.f;
      if (h < H_) {
        #pragma unroll
        for (int j = 0; j < 16; ++j) u += WkRow[h * 16 + j] * query[h * 16 + j];
      }
      Ulds[t][h] = u * 0.25f;          // 1/sqrt(dh) folded in
    }
  }
  // init softmax scratch
  {
    float* pf = &Plds[0][0];
    for (int i = t; i < 16 * 17; i += 128) pf[i] = 0.f;
    if (t < 16) scaleLds[t] = 1.0f;
  }
  __syncthreads();

  // ---------------- Phase A: flash-style single-query attention -------------
  const int hIdx = t >> 4;             // head this thread tracks (0..7)
  const int nl   = t & 15;             // n-local within tile
  float m = -__builtin_inff();
  float l = 0.f;
  v8f acc0 = {0.f, 0.f, 0.f, 0.f, 0.f, 0.f, 0.f, 0.f};   // d-cols wave*32 .. +15
  v8f acc1 = {0.f, 0.f, 0.f, 0.f, 0.f, 0.f, 0.f, 0.f};   // d-cols wave*32+16 .. +31

  // prime the pipeline: TDM load of tile 0 into buffer 0 (wave 0 issues)
  if (wave == 0)
    tdm_load_tile(embB, (unsigned)(uintptr_t)&Temb[0][0][0], N_);

  for (int tile = 0; tile < NTILE; ++tile) {
    const int n0  = tile * 16;
    const int cur = tile & 1;

    // -- wave 0: kick next tile's TDM into the other buffer, then wait for
    //    the current tile (TENSORcnt is in-order per wave).
    if (wave == 0) {
      if (tile + 1 < NTILE) {
        const int n1 = n0 + 16;
        tdm_load_tile(embB + (size_t)n1 * D_,
                      (unsigned)(uintptr_t)&Temb[cur ^ 1][0][0], N_ - n1);
        __builtin_amdgcn_s_wait_tensorcnt(1);
      } else {
        __builtin_amdgcn_s_wait_tensorcnt(0);
      }
    }
    __syncthreads();   // tile `cur` resident in LDS for all waves

    // -- scores tile S[16n x 16h] = Temb(16x128) @ U(128x16), wave 0 only
    if (wave == 0) {
      v8f sacc = {0.f, 0.f, 0.f, 0.f, 0.f, 0.f, 0.f, 0.f};
      #pragma unroll 4
      for (int kk = 0; kk < 32; ++kk) {
        const int k0 = kk * 4 + ((lane >> 4) << 1);
        v2f a; a.x = Temb[cur][lane & 15][k0]; a.y = Temb[cur][lane & 15][k0 + 1];
        v2f bb; bb.x = Ulds[k0][lane & 15]; bb.y = Ulds[k0 + 1][lane & 15];
        sacc = wmma_f32(a, bb, sacc);
      }
      const int col   = lane & 15;
      const int rbase = (lane >> 4) * 8;
      #pragma unroll
      for (int j = 0; j < 8; ++j) Slds[rbase + j][col] = sacc[j];
    }
    __syncthreads();

    // -- online softmax update (one thread per (h, n_local))
    {
      const int n  = n0 + nl;
      float sv = -__builtin_inff();
      if (n < N_ && maskB[n] == 0) sv = Slds[nl][hIdx];
      float mx = sv;
      #pragma unroll
      for (int off = 8; off; off >>= 1) mx = fmaxf(mx, __shfl_xor(mx, off, 16));
      const float nm = fmaxf(m, mx);
      const float p  = (nm == -__builtin_inff()) ? 0.f : __expf(sv - nm);
      float ps = p;
      #pragma unroll
      for (int off = 8; off; off >>= 1) ps += __shfl_xor(ps, off, 16);
      const float scale = (m == -__builtin_inff()) ? 0.f : __expf(m - nm);
      l = l * scale + ps;
      m = nm;
      Plds[nl][hIdx] = p;
      if (nl == 0) scaleLds[hIdx] = scale;   // rows 8..15 remain 1.0
    }
    __syncthreads();

    // -- rescale + accumulate: Acc[16h x 128d] += P^T(16x16) @ Temb(16x128)
    {
      const int rbase = (lane >> 4) * 8;
      #pragma unroll
      for (int j = 0; j < 8; ++j) {
        const float sj = scaleLds[rbase + j];
        acc0[j] *= sj; acc1[j] *= sj;
      }
      const int c0 = wave * 32 + (lane & 15);
      #pragma unroll
      for (int kk = 0; kk < 4; ++kk) {
        const int k0 = kk * 4 + ((lane >> 4) << 1);
        v2f a;  a.x  = Plds[k0][lane & 15];   a.y  = Plds[k0 + 1][lane & 15];
        v2f b0; b0.x = Temb[cur][k0][c0];     b0.y = Temb[cur][k0 + 1][c0];
        acc0 = wmma_f32(a, b0, acc0);
        v2f b1; b1.x = Temb[cur][k0][c0 + 16]; b1.y = Temb[cur][k0 + 1][c0 + 16];
        acc1 = wmma_f32(a, b1, acc1);
      }
    }
    __syncthreads();   // consumers done: next iteration may TDM-overwrite cur (its cur^1)
  }

  // ---------------- finalize: heads -> glimpse -> wvec ----------------------
  if (nl == 0) lLds[hIdx] = l;
  {
    const int colb  = wave * 32 + (lane & 15);
    const int rbase = (lane >> 4) * 8;
    #pragma unroll
    for (int j = 0; j < 8; ++j) {
      AccLds[rbase + j][colb]      = acc0[j];
      AccLds[rbase + j][colb + 16] = acc1[j];
    }
  }
  __syncthreads();
  {
    // heads[c] = (a_h / l_h) . Wv[:,c],  c = t, h = t>>4
    const float invl = 1.0f / lLds[hIdx];
    float acc = 0.f;
    for (int d = 0; d < D_; ++d)
      acc += AccLds[hIdx][d] * W_node[(size_t)d * (3 * D_) + D_ + t];
    headsLds[t] = acc * invl;
  }
  __syncthreads();
  {
    float g = 0.f;
    for (int c = 0; c < D_; ++c) g += headsLds[c] * W_out[c * D_ + t];
    glimpseLds[t] = g;
  }
  __syncthreads();
  {
    const float* WlRow = W_node + (size_t)t * (3 * D_) + 2 * D_;
    float wv = 0.f;
    for (int c = 0; c < D_; ++c) wv += WlRow[c] * glimpseLds[c];
    wvec[t] = wv * INVSQRT_D;
  }
  __syncthreads();

  // ---------------- Phase B: logits sweep + log_softmax ---------------------
  for (int n = t; n < N_; n += 128) {
    if (n + 128 < N_)
      __builtin_prefetch(embB + (size_t)(n + 128) * D_, 0, 0);
    const float4* er = (const float4*)(embB + (size_t)n * D_);
    float lg = 0.f;
    #pragma unroll 8
    for (int i = 0; i < 32; ++i) {
      const float4 e = er[i];
      lg += e.x * wvec[i * 4 + 0] + e.y * wvec[i * 4 + 1] +
            e.z * wvec[i * 4 + 2] + e.w * wvec[i * 4 + 3];
    }
    lg = tanhf(lg) * 10.0f;
    logitsLds[n] = maskB[n] ? -__builtin_inff() : lg;
  }
  __syncthreads();
  // block max
  float lm = -__builtin_inff();
  for (int n = t; n < N_; n += 128) lm = fmaxf(lm, logitsLds[n]);
  red[t] = lm; __syncthreads();
  for (int s = 64; s; s >>= 1) {
    if (t < s) red[t] = fmaxf(red[t], red[t + s]);
    __syncthreads();
  }
  const float gmax = red[0];
  __syncthreads();
  // block sum of exp
  float ls = 0.f;
  for (int n = t; n < N_; n += 128) {
    const float v = logitsLds[n];
    ls += (v == -__builtin_inff()) ? 0.f : __expf(v - gmax);
  }
  red[t] = ls; __syncthreads();
  for (int s = 64; s; s >>= 1) {
    if (t < s) red[t] += red[t + s];
    __syncthreads();
  }
  const float lse = gmax + __logf(red[0]);

  float* outB = out + (size_t)b * N_;
  for (int n = t; n < N_; n += 128) outB[n] = logitsLds[n] - lse;
}

extern "C" void kernel_launch(void* const* d_in, const int* in_sizes, int n_in,
                              void* d_out, int out_size, void* d_ws,
                              size_t ws_size, hipStream_t stream) {
  const float* emb     = (const float*)d_in[0];
  const int*   mask    = (const int*)d_in[1];
  const float* W_node  = (const float*)d_in[2];
  const float* W_fixed = (const float*)d_in[3];
  const float* W_step  = (const float*)d_in[4];
  const float* W_out   = (const float*)d_in[5];
  const float* W_ph    = (const float*)d_in[6];
  attn_decode_step_kernel<<<dim3(B_), dim3(128), 0, stream>>>(
      emb, mask, W_node, W_fixed, W_step, W_out, W_ph, (float*)d_out);
  (void)in_sizes; (void)n_in; (void)out_size; (void)d_ws; (void)ws_size;
}